// MixUp1D_31516470018122
// MI455X (gfx1250) — compile-verified
//
#include <hip/hip_runtime.h>
#include <cstdint>

// MixUp1D: out_x[n] = a*x[n] + (1-a)*x[perm[n]],  a = mask[n] ? lam[n] : 1.0
// N=256, C=16, L=16384  -> 262144 f32 per sample, 256 MB total.
// Memory-bound: ~768 MB traffic -> ~33 us floor at 23.3 TB/s.
// CDNA5 path: async global->LDS DMA (ASYNCcnt) double-buffered pipeline,
// b128 transfers, non-temporal b128 stores to preserve L2 for the permuted
// second read of each x chunk.

typedef float v4f __attribute__((ext_vector_type(4)));

#define SAMPLE_ELEMS (16 * 16384)   // C*L = 262144 floats
#define CHUNKS       16             // blocks per sample
#define NSTAGES      16             // pipeline stages per block
#define STAGE_ELEMS  1024           // 256 threads * 4 floats

__global__ __launch_bounds__(256)
void mixup_x_kernel(const float* __restrict__ x,
                    const int* __restrict__ perm,
                    const unsigned char* __restrict__ mask,
                    const float* __restrict__ lam,
                    float* __restrict__ out)
{
    const int n     = blockIdx.y;     // sample
    const int chunk = blockIdx.x;     // chunk within sample
    const int t     = threadIdx.x;

    // Block-uniform per-sample scalars (compiler emits s_load for these).
    const int   pn = perm[n];
    const float a  = mask[n] ? lam[n] : 1.0f;
    const float b  = 1.0f - a;        // mask==0 -> a=1, b=0 -> out == x exactly

    const size_t chunk_off = (size_t)chunk * (NSTAGES * STAGE_ELEMS);
    const float* xb  = x   + (size_t)n  * SAMPLE_ELEMS + chunk_off;
    const float* xpb = x   + (size_t)pn * SAMPLE_ELEMS + chunk_off;
    float*       ob  = out + (size_t)n  * SAMPLE_ELEMS + chunk_off;

    // Double-buffered staging: [buf][stream(x/xp)][thread] -> 16 KB LDS.
    __shared__ v4f lbuf[2][2][256];

    // Each lane DMAs one float4 per stream into its own LDS slot.
    // LDS byte offset = low 32 bits of the generic (flat) shared pointer.
    auto issue = [&](int s) {
        const int bsel = s & 1;
        uint32_t lx = (uint32_t)(uintptr_t)&lbuf[bsel][0][t];
        uint32_t lp = (uint32_t)(uintptr_t)&lbuf[bsel][1][t];
        uint64_t gx = (uint64_t)(uintptr_t)(xb  + (size_t)s * STAGE_ELEMS + (size_t)t * 4);
        uint64_t gp = (uint64_t)(uintptr_t)(xpb + (size_t)s * STAGE_ELEMS + (size_t)t * 4);
        asm volatile("global_load_async_to_lds_b128 %0, %1, off"
                     :: "v"(lx), "v"(gx) : "memory");
        asm volatile("global_load_async_to_lds_b128 %0, %1, off"
                     :: "v"(lp), "v"(gp) : "memory");
    };

    issue(0);

#pragma unroll
    for (int s = 0; s < NSTAGES; ++s) {
        if (s + 1 < NSTAGES) {
            issue(s + 1);
            // 2 outstanding async ops belong to stage s+1; stage s is done.
            asm volatile("s_wait_asynccnt 0x2" ::: "memory");
        } else {
            asm volatile("s_wait_asynccnt 0x0" ::: "memory");
        }
        const int bsel = s & 1;
        v4f vx = lbuf[bsel][0][t];   // ds_load_b128
        v4f vp = lbuf[bsel][1][t];
        v4f r;
        r.x = a * vx.x + b * vp.x;
        r.y = a * vx.y + b * vp.y;
        r.z = a * vx.z + b * vp.z;
        r.w = a * vx.w + b * vp.w;
        // Streaming output: non-temporal b128 store, keep L2 for x reuse.
        __builtin_nontemporal_store(
            r, (v4f*)(ob + (size_t)s * STAGE_ELEMS + (size_t)t * 4));
    }
}

__global__ __launch_bounds__(256)
void mixup_y_kernel(const float* __restrict__ y,
                    const int* __restrict__ perm,
                    const unsigned char* __restrict__ mask,
                    const float* __restrict__ lam,
                    float* __restrict__ outy)
{
    const int i = blockIdx.x * blockDim.x + threadIdx.x;
    if (i >= 256 * 8) return;
    const int n = i >> 3;
    const int k = i & 7;
    const int pn = perm[n];
    const float a = mask[n] ? lam[n] : 1.0f;
    outy[i] = a * y[i] + (1.0f - a) * y[pn * 8 + k];
}

extern "C" void kernel_launch(void* const* d_in, const int* in_sizes, int n_in,
                              void* d_out, int out_size, void* d_ws, size_t ws_size,
                              hipStream_t stream)
{
    (void)in_sizes; (void)n_in; (void)out_size; (void)d_ws; (void)ws_size;

    const float*         x    = (const float*)d_in[0];          // (256,16,16384) f32
    const float*         y    = (const float*)d_in[1];          // (256,8) f32
    const int*           perm = (const int*)d_in[2];            // (256,) int
    const unsigned char* mask = (const unsigned char*)d_in[3];  // (256,) bool (1B)
    const float*         lam  = (const float*)d_in[4];          // (256,) f32

    float* out_x = (float*)d_out;                               // aug_x first
    float* out_y = out_x + (size_t)256 * SAMPLE_ELEMS;          // then aug_y

    dim3 grid(CHUNKS, 256);
    mixup_x_kernel<<<grid, 256, 0, stream>>>(x, perm, mask, lam, out_x);
    mixup_y_kernel<<<(2048 + 255) / 256, 256, 0, stream>>>(y, perm, mask, lam, out_y);
}